// SparseKernelConv_8065948582434
// MI455X (gfx1250) — compile-verified
//
#include <hip/hip_runtime.h>

// ext-vector types for WMMA fragments (fp32 path: A/B = 2 VGPRs, C/D = 8 VGPRs, wave32)
typedef float v2f __attribute__((ext_vector_type(2)));
typedef float v8f __attribute__((ext_vector_type(8)));

#define BKEY 130            // key base (GRID + 2)
#define KNBR 27             // 3x3x3 taps
#define CH 32               // Cin == Cout == Ccond
#define PTS_PER_BLOCK 128   // 8 waves x 16 points
#define THREADS 256
#define LDS_STRIDE 33       // pad to avoid LDS bank conflicts

// ---- kernel 1: invert permutation  stoo[ip[i]] = i ----
__global__ void stoo_kernel(const int* __restrict__ ip, int* __restrict__ stoo, int M) {
    int i = blockIdx.x * blockDim.x + threadIdx.x;
    if (i < M) stoo[ip[i]] = i;
}

// ---- kernel 2: sparse conv + FiLM + residual, WMMA f32 16x16x4 ----
__global__ __launch_bounds__(THREADS)
void spconv_film_kernel(const float* __restrict__ feats,
                        const int*   __restrict__ coords,
                        const int*   __restrict__ sk,
                        const int*   __restrict__ stoo,
                        const float* __restrict__ cond,
                        const int*   __restrict__ offsets,
                        const float* __restrict__ weight,
                        const float* __restrict__ bias,
                        const float* __restrict__ gamma_w,
                        const float* __restrict__ gamma_b,
                        const float* __restrict__ beta_w,
                        const float* __restrict__ beta_b,
                        float* __restrict__ out,
                        int M) {
    __shared__ int   sIdx[PTS_PER_BLOCK * KNBR];   // gather index per (point, tap); -1 = invalid
    __shared__ float sW [CH * LDS_STRIDE];         // current tap's 32x32 weight
    __shared__ float sGw[CH * LDS_STRIDE];         // FiLM gamma weight
    __shared__ float sBw[CH * LDS_STRIDE];         // FiLM beta weight

    const int tid = threadIdx.x;
    const int blockBase = blockIdx.x * PTS_PER_BLOCK;

    // stage FiLM weights once ([c][o], padded stride)
    for (int i = tid; i < CH * CH; i += THREADS) {
        int c = i >> 5, o = i & 31;
        sGw[c * LDS_STRIDE + o] = gamma_w[i];
        sBw[c * LDS_STRIDE + o] = beta_w[i];
    }

    // ---- phase 1: block-cooperative neighbor search (lower_bound on sk) ----
    for (int e = tid; e < PTS_PER_BLOCK * KNBR; e += THREADS) {
        int ml = e / KNBR;
        int k  = e - ml * KNBR;
        int m  = blockBase + ml;
        int idx = -1;
        if (m < M) {
            int cx = coords[m * 3 + 0] + offsets[k * 3 + 0] + 1;
            int cy = coords[m * 3 + 1] + offsets[k * 3 + 1] + 1;
            int cz = coords[m * 3 + 2] + offsets[k * 3 + 2] + 1;
            int q  = (cx * BKEY + cy) * BKEY + cz;
            int lo = 0, hi = M;
            while (lo < hi) {                    // searchsorted side='left'
                int mid = (lo + hi) >> 1;
                if (sk[mid] < q) lo = mid + 1; else hi = mid;
            }
            int posc = lo < (M - 1) ? lo : (M - 1);
            if (sk[posc] == q) idx = stoo[posc];
        }
        sIdx[e] = idx;
    }
    __syncthreads();

    // ---- phase 2: per-wave 16-point tile, out[16,32] via WMMA f32 16x16x4 ----
    const int wave = tid >> 5;
    const int lane = tid & 31;
    const int n0   = lane & 15;   // N index (and M index for A fragments)
    const int half = lane >> 4;   // K-half selector inside 4-wide K window
    const int mBase = blockBase + wave * 16;

    float b0v = bias[n0], b1v = bias[16 + n0];
    v8f cacc0 = { b0v, b0v, b0v, b0v, b0v, b0v, b0v, b0v };   // N = 0..15 tile
    v8f cacc1 = { b1v, b1v, b1v, b1v, b1v, b1v, b1v, b1v };   // N = 16..31 tile

    const int* myIdx = &sIdx[(wave * 16 + n0) * KNBR];

    for (int k = 0; k < KNBR; ++k) {
        __syncthreads();                         // previous sW consumers done
        const float* wk = weight + k * (CH * CH);
        for (int i = tid; i < CH * CH; i += THREADS) {
            int c = i >> 5, o = i & 31;
            sW[c * LDS_STRIDE + o] = wk[i];
        }
        __syncthreads();

        int idx = myIdx[k];
        const float* arow = feats + (long)(idx < 0 ? 0 : idx) * CH;
        float amask = idx < 0 ? 0.0f : 1.0f;     // zero invalid neighbor row (matches reference)

        #pragma unroll
        for (int cc = 0; cc < 8; ++cc) {
            int c0 = cc * 4 + half * 2;
            v2f a;  a.x = arow[c0] * amask;  a.y = arow[c0 + 1] * amask;
            v2f b0, b1;
            b0.x = sW[c0 * LDS_STRIDE + n0];        b0.y = sW[(c0 + 1) * LDS_STRIDE + n0];
            b1.x = sW[c0 * LDS_STRIDE + 16 + n0];   b1.y = sW[(c0 + 1) * LDS_STRIDE + 16 + n0];
            cacc0 = __builtin_amdgcn_wmma_f32_16x16x4_f32(false, a, false, b0, (short)0, cacc0, false, false);
            cacc1 = __builtin_amdgcn_wmma_f32_16x16x4_f32(false, a, false, b1, (short)0, cacc1, false, false);
        }
    }

    // ---- phase 3: FiLM gamma/beta via the same WMMA tiling (A = cond rows) ----
    float g0v = gamma_b[n0], g1v = gamma_b[16 + n0];
    float t0v = beta_b[n0],  t1v = beta_b[16 + n0];
    v8f gacc0 = { g0v, g0v, g0v, g0v, g0v, g0v, g0v, g0v };
    v8f gacc1 = { g1v, g1v, g1v, g1v, g1v, g1v, g1v, g1v };
    v8f bacc0 = { t0v, t0v, t0v, t0v, t0v, t0v, t0v, t0v };
    v8f bacc1 = { t1v, t1v, t1v, t1v, t1v, t1v, t1v, t1v };

    int mA = mBase + n0;
    const float* crow = cond + (long)(mA < M ? mA : 0) * CH;
    float cmask = (mA < M) ? 1.0f : 0.0f;

    #pragma unroll
    for (int cc = 0; cc < 8; ++cc) {
        int c0 = cc * 4 + half * 2;
        v2f a;  a.x = crow[c0] * cmask;  a.y = crow[c0 + 1] * cmask;
        v2f g0, g1, t0, t1;
        g0.x = sGw[c0 * LDS_STRIDE + n0];       g0.y = sGw[(c0 + 1) * LDS_STRIDE + n0];
        g1.x = sGw[c0 * LDS_STRIDE + 16 + n0];  g1.y = sGw[(c0 + 1) * LDS_STRIDE + 16 + n0];
        t0.x = sBw[c0 * LDS_STRIDE + n0];       t0.y = sBw[(c0 + 1) * LDS_STRIDE + n0];
        t1.x = sBw[c0 * LDS_STRIDE + 16 + n0];  t1.y = sBw[(c0 + 1) * LDS_STRIDE + 16 + n0];
        gacc0 = __builtin_amdgcn_wmma_f32_16x16x4_f32(false, a, false, g0, (short)0, gacc0, false, false);
        gacc1 = __builtin_amdgcn_wmma_f32_16x16x4_f32(false, a, false, g1, (short)0, gacc1, false, false);
        bacc0 = __builtin_amdgcn_wmma_f32_16x16x4_f32(false, a, false, t0, (short)0, bacc0, false, false);
        bacc1 = __builtin_amdgcn_wmma_f32_16x16x4_f32(false, a, false, t1, (short)0, bacc1, false, false);
    }

    // ---- combine: gamma*conv + beta + residual; C/D layout: vgpr r -> M = r + half*8, lane -> N ----
    #pragma unroll
    for (int r = 0; r < 8; ++r) {
        int m = mBase + r + half * 8;
        if (m < M) {
            long base = (long)m * CH;
            float f0 = feats[base + n0];
            float f1 = feats[base + 16 + n0];
            out[base + n0]      = gacc0[r] * cacc0[r] + bacc0[r] + f0;
            out[base + 16 + n0] = gacc1[r] * cacc1[r] + bacc1[r] + f1;
        }
    }
}

extern "C" void kernel_launch(void* const* d_in, const int* in_sizes, int n_in,
                              void* d_out, int out_size, void* d_ws, size_t ws_size,
                              hipStream_t stream) {
    const float* feats   = (const float*)d_in[0];
    const int*   coords  = (const int*)  d_in[1];
    const int*   sk      = (const int*)  d_in[2];
    const int*   ip      = (const int*)  d_in[3];
    const float* cond    = (const float*)d_in[4];
    const int*   offsets = (const int*)  d_in[5];
    const float* weight  = (const float*)d_in[6];
    const float* bias    = (const float*)d_in[7];
    const float* gamma_w = (const float*)d_in[8];
    const float* gamma_b = (const float*)d_in[9];
    const float* beta_w  = (const float*)d_in[10];
    const float* beta_b  = (const float*)d_in[11];

    const int M = in_sizes[0] / CH;
    int* stoo = (int*)d_ws;   // M ints of scratch

    stoo_kernel<<<(M + 255) / 256, 256, 0, stream>>>(ip, stoo, M);

    int nblk = (M + PTS_PER_BLOCK - 1) / PTS_PER_BLOCK;
    spconv_film_kernel<<<nblk, THREADS, 0, stream>>>(
        feats, coords, sk, stoo, cond, offsets, weight, bias,
        gamma_w, gamma_b, beta_w, beta_b, (float*)d_out, M);
}